// KNN_48146583388931
// MI455X (gfx1250) — compile-verified
//
#include <hip/hip_runtime.h>
#include <math.h>

typedef float v2f __attribute__((ext_vector_type(2)));
typedef float v4f __attribute__((ext_vector_type(4)));
typedef float v8f __attribute__((ext_vector_type(8)));

#define B_   4
#define N_   8192
#define M_   4096
#define K_   16
#define WAVES_PER_BLOCK 4
#define PITCH 20   // floats; 20*L mod 64 distinct for L=0..15, rows 16B-aligned

struct WaveLds {
    float d[16 * PITCH];      // 16x16 D tile
    float q2[16];             // ||q||^2 per query in tile
    float mdist[2 * 16 * 16]; // two half top-k lists per query (merge stage)
    int   midx[2 * 16 * 16];
};

__global__ __launch_bounds__(128)
void knn_wmma_kernel(const float* __restrict__ ref,
                     const float* __restrict__ query,
                     float* __restrict__ outD,
                     float* __restrict__ outI) {
    __shared__ WaveLds lds[WAVES_PER_BLOCK];

    const int lane = threadIdx.x & 31;
    const int wave = threadIdx.x >> 5;
    WaveLds& L = lds[wave];

    const int tile = blockIdx.x * WAVES_PER_BLOCK + wave;   // global 16-query tile
    const int b    = tile / (M_ / 16);
    const int qt   = tile % (M_ / 16);

    const int  q    = lane & 15;           // query (row) this lane serves
    const bool hi   = (lane >= 16);
    const int  hi8  = hi ? 8 : 0;

    // ---- Query point for this lane's row of A ----
    const int    mq = qt * 16 + q;
    const float* qp = query + ((size_t)b * M_ + mq) * 3;
    const float qx = qp[0], qy = qp[1], qz = qp[2];
    const float q2 = qx * qx + qy * qy + qz * qz;

    // A(16x4): row m = [-2qx,-2qy,-2qz, 1]; lanes0-15 hold K0,K1; lanes16-31 K2,K3
    v2f am;
    am.x = hi ? (-2.0f * qz) : (-2.0f * qx);
    am.y = hi ? 1.0f         : (-2.0f * qy);

    if (!hi) L.q2[q] = q2;
    asm volatile("s_wait_dscnt 0" ::: "memory");
    __builtin_amdgcn_wave_barrier();

    // C-init: row m = v + hi8 needs q2[m]  (kept in registers across the loop)
    float q2row[8];
#pragma unroll
    for (int v = 0; v < 8; ++v) q2row[v] = L.q2[v + hi8];
    __builtin_amdgcn_wave_barrier();

    // ---- Per-lane sorted ascending top-16 (bd[15] = current worst) ----
    float bd[16];
    int   bi[16];
#pragma unroll
    for (int u = 0; u < 16; ++u) { bd[u] = 3.402823466e+38f; bi[u] = 0; }

    // Guarded bubble-insert, branch-free inside (cndmask only)
    auto ins1 = [&](float raw, int gid) {
        float d2  = fmaxf(raw, 0.0f);
        bool  ins = d2 < bd[15];
        bd[15] = ins ? d2  : bd[15];
        bi[15] = ins ? gid : bi[15];
#pragma unroll
        for (int u = 15; u >= 1; --u) {
            bool  sw = bd[u] < bd[u - 1];
            float dl = sw ? bd[u]     : bd[u - 1];
            float dh = sw ? bd[u - 1] : bd[u];
            int   il = sw ? bi[u]     : bi[u - 1];
            int   ih = sw ? bi[u - 1] : bi[u];
            bd[u - 1] = dl; bd[u] = dh;
            bi[u - 1] = il; bi[u] = ih;
        }
    };

    const float* refb = ref + (size_t)b * N_ * 3;

    for (int j = 0; j < N_ / 16; ++j) {
        if (j + 2 < N_ / 16)
            __builtin_prefetch(refb + (size_t)((j + 2) * 16 + q) * 3, 0, 3);

        // B(4x16): col n = [rx, ry, rz, ||r||^2]^T, branchless lane split
        const float* rp = refb + (size_t)(j * 16 + q) * 3;
        const float rx = rp[0], ry = rp[1], rz = rp[2];
        const float r2 = rx * rx + ry * ry + rz * rz;
        v2f bm;
        bm.x = hi ? rz : rx;
        bm.y = hi ? r2 : ry;

        v8f c;
#pragma unroll
        for (int v = 0; v < 8; ++v) c[v] = q2row[v];

        // d2 = q2 + (-2 q . r + r2): one fp32 WMMA per 16x16 tile
        c = __builtin_amdgcn_wmma_f32_16x16x4_f32(
                false, am, false, bm, (short)0, c, false, false);

        // Scatter D to wave-private LDS: row m = v + hi8, col = q
#pragma unroll
        for (int v = 0; v < 8; ++v)
            L.d[(v + hi8) * PITCH + q] = c[v];
        asm volatile("s_wait_dscnt 0" ::: "memory");
        __builtin_amdgcn_wave_barrier();

        // Fetch this lane's 8 candidates as two b128 loads
        const v4f* row = (const v4f*)&L.d[q * PITCH + hi8];
        v4f e0 = row[0];
        v4f e1 = row[1];

        // 8-way min tree -> single steady-state threshold test
        float m0 = fminf(fminf(e0.x, e0.y), fminf(e0.z, e0.w));
        float m1 = fminf(fminf(e1.x, e1.y), fminf(e1.z, e1.w));
        float cm = fmaxf(fminf(m0, m1), 0.0f);

        if (cm < bd[15]) {               // rare after warm-up
            const int gb = j * 16 + hi8;
            ins1(e0.x, gb + 0); ins1(e0.y, gb + 1);
            ins1(e0.z, gb + 2); ins1(e0.w, gb + 3);
            ins1(e1.x, gb + 4); ins1(e1.y, gb + 5);
            ins1(e1.z, gb + 6); ins1(e1.w, gb + 7);
        }
        __builtin_amdgcn_wave_barrier();
    }

    // ---- Merge the two half-lists per query through LDS ----
#pragma unroll
    for (int u = 0; u < 16; ++u) {
        L.mdist[((hi ? 16 : 0) + q) * 16 + u] = bd[u];
        L.midx [((hi ? 16 : 0) + q) * 16 + u] = bi[u];
    }
    asm volatile("s_wait_dscnt 0" ::: "memory");
    __builtin_amdgcn_wave_barrier();

    if (!hi) {
        const float* A  = &L.mdist[(0 + q) * 16];
        const int*   AI = &L.midx [(0 + q) * 16];
        const float* Bv = &L.mdist[(16 + q) * 16];
        const int*   BI = &L.midx [(16 + q) * 16];
        int ia = 0, ib = 0;
        const size_t base = ((size_t)b * M_ + mq) * K_;
        for (int u = 0; u < K_; ++u) {
            float da = A[ia],  db = Bv[ib];
            int   xa = AI[ia], xb = BI[ib];
            bool takeA = (da < db) || (da == db && xa <= xb);
            float dsel = takeA ? da : db;
            int   isel = takeA ? xa : xb;
            ia += takeA; ib += !takeA;
            outD[base + u] = sqrtf(dsel);
            outI[base + u] = (float)isel;
        }
    }
}

extern "C" void kernel_launch(void* const* d_in, const int* in_sizes, int n_in,
                              void* d_out, int out_size, void* d_ws, size_t ws_size,
                              hipStream_t stream) {
    (void)in_sizes; (void)n_in; (void)d_ws; (void)ws_size; (void)out_size;
    const float* ref   = (const float*)d_in[0];
    const float* query = (const float*)d_in[1];
    float* outD = (float*)d_out;
    float* outI = outD + (size_t)B_ * M_ * K_;

    const int tiles  = B_ * (M_ / 16);           // 1024 query tiles, 1 wave each
    const int blocks = tiles / WAVES_PER_BLOCK;  // 256 blocks x 128 threads
    knn_wmma_kernel<<<blocks, 128, 0, stream>>>(ref, query, outD, outI);
}